// GLNSampler_BC_352187318915
// MI455X (gfx1250) — compile-verified
//
#include <hip/hip_runtime.h>

#define NN 4096
#define DD 256
#define MM 5
#define K_INTRA 20
#define K_INTER 10
#define KNN 30
#define NEGF (-1e30f)
#define WORDS_PER_ROW (NN / 32) // 128

typedef __attribute__((ext_vector_type(2))) float v2f;
typedef __attribute__((ext_vector_type(8))) float v8f;

// ---------------------------------------------------------------------------
// Kernel 1: sim = E @ E^T via V_WMMA_F32_16X16X4_F32.
// Each wave computes a 16x64 strip (4 accumulators reuse the A fragment).
// A 16x4 f32 fragment: lane<16 -> (K=0,1), lane>=16 -> (K=2,3), row = M=lane%16.
// B 4x16 fragment for E^T tile has the identical load pattern (row = N=lane%16).
// ---------------------------------------------------------------------------
__global__ __launch_bounds__(256) void gemm_sim_kernel(const float* __restrict__ E,
                                                       float* __restrict__ sim) {
  const int lane = threadIdx.x & 31;
  const int wave = threadIdx.x >> 5;
  const int lr = lane & 15;
  const int hi = lane >> 4; // 0 or 1
  const int row0 = (blockIdx.y * 8 + wave) * 16;
  const int col0 = blockIdx.x * 64;

  const float* arow  = E + (size_t)(row0 + lr) * DD + 2 * hi;
  const float* brow0 = E + (size_t)(col0 +  0 + lr) * DD + 2 * hi;
  const float* brow1 = E + (size_t)(col0 + 16 + lr) * DD + 2 * hi;
  const float* brow2 = E + (size_t)(col0 + 32 + lr) * DD + 2 * hi;
  const float* brow3 = E + (size_t)(col0 + 48 + lr) * DD + 2 * hi;

  v8f acc0 = {}, acc1 = {}, acc2 = {}, acc3 = {};
  for (int k = 0; k < DD; k += 4) {
    v2f a  = *(const v2f*)(arow + k);
    v2f b0 = *(const v2f*)(brow0 + k);
    v2f b1 = *(const v2f*)(brow1 + k);
    v2f b2 = *(const v2f*)(brow2 + k);
    v2f b3 = *(const v2f*)(brow3 + k);
    acc0 = __builtin_amdgcn_wmma_f32_16x16x4_f32(false, a, false, b0, (short)0, acc0, false, false);
    acc1 = __builtin_amdgcn_wmma_f32_16x16x4_f32(false, a, false, b1, (short)0, acc1, false, false);
    acc2 = __builtin_amdgcn_wmma_f32_16x16x4_f32(false, a, false, b2, (short)0, acc2, false, false);
    acc3 = __builtin_amdgcn_wmma_f32_16x16x4_f32(false, a, false, b3, (short)0, acc3, false, false);
  }

  // C/D layout: lane<16 -> M=v, N=lane; lane>=16 -> M=8+v, N=lane-16.
#pragma unroll
  for (int v = 0; v < 8; ++v) {
    const int r = row0 + hi * 8 + v;
    float* o = sim + (size_t)r * NN + col0 + lr;
    o[0]  = acc0[v];
    o[16] = acc1[v];
    o[32] = acc2[v];
    o[48] = acc3[v];
  }
}

// ---------------------------------------------------------------------------
// Kernel 2: per-row masked top-k. One 256-thread block per row; row cached in
// LDS. Iterative argmax with smallest-index tie-break (matches jax top_k).
// Phase 0: intra (same batch), 21 picks, store picks 1..20. Phase 1: inter, 10.
// ---------------------------------------------------------------------------
__global__ __launch_bounds__(256) void topk_kernel(const float* __restrict__ sim,
                                                   const int* __restrict__ batch_id,
                                                   int* __restrict__ knn) {
  __shared__ float vals[NN];
  __shared__ float wmax[8];
  __shared__ int widx[8];

  const int i = blockIdx.x;
  const int tid = threadIdx.x;
  const int lane = tid & 31;
  const int wave = tid >> 5;
  const int bi = batch_id[i];
  const float* row = sim + (size_t)i * NN;

  for (int phase = 0; phase < 2; ++phase) {
    for (int j = tid; j < NN; j += 256) {
      const bool samegrp = (batch_id[j] == bi);
      const float s = row[j];
      vals[j] = (samegrp == (phase == 0)) ? s : NEGF;
    }
    __syncthreads();

    const int iters = (phase == 0) ? (K_INTRA + 1) : K_INTER;
    for (int t = 0; t < iters; ++t) {
      float bv = NEGF;
      int bj = NN;
      for (int j = tid; j < NN; j += 256) {
        const float v = vals[j];
        if (v > bv) { bv = v; bj = j; } // strict > keeps smallest index per-thread
      }
      // wave32 shuffle reduction, smallest index on ties
      for (int off = 16; off > 0; off >>= 1) {
        const float ov = __shfl_down(bv, off);
        const int oj = __shfl_down(bj, off);
        if (ov > bv || (ov == bv && oj < bj)) { bv = ov; bj = oj; }
      }
      if (lane == 0) { wmax[wave] = bv; widx[wave] = bj; }
      __syncthreads();
      if (tid == 0) {
        float fv = wmax[0];
        int fj = widx[0];
        for (int w = 1; w < 8; ++w)
          if (wmax[w] > fv || (wmax[w] == fv && widx[w] < fj)) { fv = wmax[w]; fj = widx[w]; }
        vals[fj] = NEGF;
        if (phase == 0) {
          if (t > 0) knn[(size_t)i * KNN + (t - 1)] = fj; // drop rank-0 pick
        } else {
          knn[(size_t)i * KNN + K_INTRA + t] = fj;
        }
      }
      __syncthreads();
    }
  }
}

// ---------------------------------------------------------------------------
// Kernels 3a/3b: kNN membership bitset (N x N bits = 2 MB, L2-resident).
// ---------------------------------------------------------------------------
__global__ void zero_bits_kernel(unsigned int* __restrict__ bits, int nwords) {
  const int idx = blockIdx.x * blockDim.x + threadIdx.x;
  if (idx < nwords) bits[idx] = 0u;
}

__global__ void set_bits_kernel(const int* __restrict__ knn,
                                unsigned int* __restrict__ bits) {
  const int idx = blockIdx.x * blockDim.x + threadIdx.x;
  if (idx >= NN * KNN) return;
  const int i = idx / KNN;
  const int j = knn[idx];
  atomicOr(&bits[i * WORDS_PER_ROW + (j >> 5)], 1u << (j & 31));
}

// ---------------------------------------------------------------------------
// Kernel 4: locality[i,j] = (i!=j && j in knn(i) && i in knn(j)) ? adj[i,j] : 0
// Pure streaming on adj/locality; bitset probes hit L2.
// ---------------------------------------------------------------------------
__global__ __launch_bounds__(256) void locality_kernel(const float* __restrict__ adj,
                                                       const unsigned int* __restrict__ bits,
                                                       float* __restrict__ out) {
  const size_t idx = (size_t)blockIdx.x * blockDim.x + threadIdx.x;
  if (idx >= (size_t)NN * NN) return;
  const int i = (int)(idx >> 12);       // / 4096
  const int j = (int)(idx & (NN - 1));  // % 4096
  const unsigned int bij = (bits[i * WORDS_PER_ROW + (j >> 5)] >> (j & 31)) & 1u;
  const unsigned int bji = (bits[j * WORDS_PER_ROW + (i >> 5)] >> (i & 31)) & 1u;
  out[idx] = (i != j && bij && bji) ? adj[idx] : 0.0f;
}

// ---------------------------------------------------------------------------
// Kernel 5: all_close[i,t] = mutual(i, knn[i,t]) || all_m(labels agree)
// ---------------------------------------------------------------------------
__global__ void allclose_kernel(const int* __restrict__ knn,
                                const unsigned int* __restrict__ bits,
                                const int* __restrict__ labels,
                                float* __restrict__ out2) {
  const int idx = blockIdx.x * blockDim.x + threadIdx.x;
  if (idx >= NN * KNN) return;
  const int i = idx / KNN;
  const int j = knn[idx];
  const unsigned int bij = (bits[i * WORDS_PER_ROW + (j >> 5)] >> (j & 31)) & 1u;
  const unsigned int bji = (bits[j * WORDS_PER_ROW + (i >> 5)] >> (i & 31)) & 1u;
  bool ok = (i != j) && bij && bji; // mutual
  if (!ok) {
    ok = true;
#pragma unroll
    for (int m = 0; m < MM; ++m) {
      if (labels[m * NN + i] != labels[m * NN + j]) { ok = false; break; }
    }
  }
  out2[idx] = ok ? 1.0f : 0.0f;
}

extern "C" void kernel_launch(void* const* d_in, const int* in_sizes, int n_in,
                              void* d_out, int out_size, void* d_ws, size_t ws_size,
                              hipStream_t stream) {
  const float* adj = (const float*)d_in[0];
  const float* enc = (const float*)d_in[1];
  const int* batch_id = (const int*)d_in[2];
  const int* labels = (const int*)d_in[3];
  (void)in_sizes; (void)n_in; (void)out_size; (void)ws_size;

  char* ws = (char*)d_ws;
  float* sim = (float*)ws;                                               // 64 MB
  int* knn = (int*)(ws + (size_t)NN * NN * sizeof(float));               // 480 KB
  unsigned int* bits =
      (unsigned int*)(ws + (size_t)NN * NN * sizeof(float) + (size_t)NN * KNN * sizeof(int)); // 2 MB

  float* locality = (float*)d_out;
  float* allclose = (float*)d_out + (size_t)NN * NN;

  gemm_sim_kernel<<<dim3(NN / 64, NN / (16 * 8)), 256, 0, stream>>>(enc, sim);
  topk_kernel<<<NN, 256, 0, stream>>>(sim, batch_id, knn);
  zero_bits_kernel<<<(NN * WORDS_PER_ROW + 255) / 256, 256, 0, stream>>>(bits, NN * WORDS_PER_ROW);
  set_bits_kernel<<<(NN * KNN + 255) / 256, 256, 0, stream>>>(knn, bits);
  locality_kernel<<<(unsigned)(((size_t)NN * NN + 255) / 256), 256, 0, stream>>>(adj, bits, locality);
  allclose_kernel<<<(NN * KNN + 255) / 256, 256, 0, stream>>>(knn, bits, labels, allclose);
}